// TtMistralAttention_78408922956349
// MI455X (gfx1250) — compile-verified
//
#include <hip/hip_runtime.h>
#include <hip/hip_bf16.h>
#include <math.h>
#include <stdint.h>

// ---------------------------------------------------------------------------
// Mistral-style single-token attention decode for MI455X (gfx1250, wave32).
// fp32 end-to-end; matrix work on V_WMMA_F32_16X16X4_F32; q staging via
// GLOBAL_LOAD_ASYNC_TO_LDS_B128 (ASYNCcnt). Workload is memory-bound
// (~300MB @ 23.3TB/s ≈ 13us), so loops are structured as pure
// load+wmma+induction streams with no per-iteration address math.
// ---------------------------------------------------------------------------

typedef __attribute__((ext_vector_type(2))) float v2f;
typedef __attribute__((ext_vector_type(8))) float v8f;

#define WMMA_F32X4(acc, a, b) \
  __builtin_amdgcn_wmma_f32_16x16x4_f32(false, (a), false, (b), (short)0, (acc), false, false)

// CDNA5 async global->LDS: operand0 = per-lane LDS byte offset (VDST),
// operand1 = 64-bit global address (VADDR), GV mode. Tracked by ASYNCcnt.
#define ASYNC_GLB_TO_LDS_B128(ldsoff, gaddr)                                   \
  asm volatile("global_load_async_to_lds_b128 %0, %1, off"                     \
               :: "v"(ldsoff), "v"(gaddr) : "memory")
#define WAIT_ASYNCCNT0() asm volatile("s_wait_asynccnt 0x0" ::: "memory")

#define DIMv    4096
#define NHv     32
#define NKVv    8
#define HDv     128
#define Bv      8
#define PADLEN  2048
#define STARTP  2047
#define SWv     4096

static __device__ __forceinline__ int imin(int a, int b) { return a < b ? a : b; }

// ---------------------------------------------------------------------------
// Generic 16x16-tile fp32 WMMA GEMM:  C[M,N] = alpha * (A[M,K] @ B[K,N])
// One wave per tile. Padding rows are CLAMPED (duplicate row M-1) instead of
// zeroed: D row m depends only on A row m, and rows >= M are never stored,
// so no predication is needed and EXEC stays all-1s for every WMMA.
// K must be a multiple of 16 (4096 / 128 here).
//
// VGPR layouts per CDNA5 ISA 7.12.2 (wave32):
//   A 16x4 f32 : lane m = lane&15 holds row m; VGPR j holds K = (lane>>4)*2+j
//   B 4x16 f32 : lane n = lane&15 holds col n; VGPR j holds K = (lane>>4)*2+j
//   C/D 16x16  : VGPR v: lanes 0-15 -> row v, lanes 16-31 -> row v+8
// ---------------------------------------------------------------------------
__global__ __launch_bounds__(32)
void gemm16_f32_wmma(const float* __restrict__ A, const float* __restrict__ B,
                     float* __restrict__ C, int M, int N, int K,
                     long sAb, long sBb, long sCb, float alpha)
{
  const int lane = threadIdx.x & 31;
  const int nt = blockIdx.x;
  const int mt = blockIdx.y;
  const int bz = blockIdx.z;
  A += (long)bz * sAb;
  B += (long)bz * sBb;
  C += (long)bz * sCb;

  const int mr = imin(mt * 16 + (lane & 15), M - 1);   // clamped source row
  const int n  = nt * 16 + (lane & 15);
  const int k0 = (lane >> 4) << 1;                     // 0 or 2
  const float* Arow = A + (long)mr * K;

  // 32-bit element-offset induction: saddr(B) + voffset addressing, no muls.
  int boff = k0 * N + n;                               // B[k0][n]
  const int bstep = 4 * N;
  const int pf    = 16 * N;

  v8f acc = {};
  for (int kb = 0; kb < K; kb += 16) {
    __builtin_prefetch(B + boff + pf, 0, 1);           // global_prefetch_b8, ~16 rows ahead
#pragma unroll
    for (int u = 0; u < 4; ++u) {
      v2f a = *(const v2f*)(Arow + kb + 4 * u + k0);   // 8B aligned (k even)
      v2f b;
      b[0] = B[boff];
      b[1] = B[boff + N];
      boff += bstep;
      acc = WMMA_F32X4(acc, a, b);
    }
  }

  const int rbase = (lane < 16) ? 0 : 8;
  const int cn    = nt * 16 + (lane & 15);
#pragma unroll
  for (int v = 0; v < 8; ++v) {
    const int row = mt * 16 + rbase + v;
    if (row < M) C[(long)row * N + cn] = acc[v] * alpha;
  }
}

// ---------------------------------------------------------------------------
// Fused attention per (b, kv-group): scores -> softmax -> probs @ V.
// 256 threads (8 waves) per (b,g). Scores (4 q-heads x 2048 positions) live
// in LDS with padded strides (132 / 2064 floats) for bank-conflict-free rows.
// Cache row STARTP is substituted with the fresh k/v (functional update; the
// input caches are never written).
// ---------------------------------------------------------------------------
__global__ __launch_bounds__(256)
void attn_decode(const float* __restrict__ q,         // [B, NH, HD], pre-scaled by 1/sqrt(HD)
                 const float* __restrict__ knew,      // [B, NKV, HD] (rotated)
                 const float* __restrict__ vnew,      // [B, NKV, HD]
                 const float* __restrict__ cache_k,   // [B, NKV, SW, HD]
                 const float* __restrict__ cache_v,   // [B, NKV, SW, HD]
                 const float* __restrict__ attn_mask, // [NH, B, PADLEN]
                 float* __restrict__ out)             // [B, NH*HD]
{
  __shared__ float qs[4][HDv + 4];      // stride 132
  __shared__ float sc[4][PADLEN + 16];  // stride 2064 (row offset = 16 banks)
  __shared__ float red[4][64];
  __shared__ float rowstat[4];

  const int tid  = threadIdx.x;
  const int lane = tid & 31;
  const int wave = tid >> 5;            // 0..7
  const int b    = blockIdx.x >> 3;
  const int g    = blockIdx.x & 7;
  const int k0   = (lane >> 4) << 1;    // 0 or 2

  // ---- Stage the 4-head q tile via CDNA5 async global->LDS DMA ----
  // Waves 0..3 each move one 128-float row: lane L copies 16B chunk L.
  if (wave < 4) {
    const float* gsrc = q + ((long)b * NHv + g * 4 + wave) * HDv + lane * 4;
    const unsigned ldsoff = (unsigned)(uintptr_t)&qs[wave][lane * 4];
    ASYNC_GLB_TO_LDS_B128(ldsoff, (unsigned long long)(uintptr_t)gsrc);
  }
  WAIT_ASYNCCNT0();
  __syncthreads();

  const float* kbase    = cache_k + ((long)(b * NKVv + g)) * SWv * HDv;
  const float* krow_new = knew + (long)(b * NKVv + g) * HDv;

  // Hoist the A fragment (q rows, clamped lane&3) out of the tile loop:
  // 32 v2f = the whole 16x128 A operand for every score tile.
  const int qr = lane & 3;
  v2f aq[32];
#pragma unroll
  for (int u = 0; u < 32; ++u)
    aq[u] = *(const v2f*)&qs[qr][u * 4 + k0];

  // ---- Phase 1: scores[4, 2048] = q_g @ K^T ----
  for (int t = wave; t < PADLEN / 16; t += 8) {
    const int pos = t * 16 + (lane & 15);
    const float* krow = (pos == STARTP) ? krow_new : (kbase + (long)pos * HDv);
    v8f acc = {};
#pragma unroll
    for (int u = 0; u < 32; ++u) {
      v2f bb = *(const v2f*)(krow + u * 4 + k0);  // imm-offset global_load_b64
      acc = WMMA_F32X4(acc, aq[u], bb);
    }
    if (lane < 16) {
#pragma unroll
      for (int v = 0; v < 4; ++v) sc[v][t * 16 + lane] = acc[v];
    }
  }
  __syncthreads();

  // ---- Phase 2: softmax over 2048 per q-row (mask applied; zero here) ----
  const int r  = tid >> 6;   // 0..3
  const int ix = tid & 63;
  const float* mrow = attn_mask + ((long)(g * 4 + r) * Bv + b) * PADLEN;

  float mx = -3.4e38f;
#pragma unroll 4
  for (int j = 0; j < PADLEN / 64; ++j) {
    const int l = ix + 64 * j;
    const float s = sc[r][l] + mrow[l];
    sc[r][l] = s;
    mx = fmaxf(mx, s);
  }
  red[r][ix] = mx;
  __syncthreads();
  if (tid < 4) {
    float m2 = -3.4e38f;
    for (int j = 0; j < 64; ++j) m2 = fmaxf(m2, red[tid][j]);
    rowstat[tid] = m2;
  }
  __syncthreads();
  const float rmax = rowstat[r];
  float sum = 0.0f;
#pragma unroll 4
  for (int j = 0; j < PADLEN / 64; ++j) {
    const int l = ix + 64 * j;
    const float e = __expf(sc[r][l] - rmax);
    sc[r][l] = e;
    sum += e;
  }
  red[r][ix] = sum;
  __syncthreads();
  if (tid < 4) {
    float s2 = 0.0f;
    for (int j = 0; j < 64; ++j) s2 += red[tid][j];
    rowstat[tid] = 1.0f / s2;
  }
  __syncthreads();
  const float rinv = rowstat[r];
#pragma unroll 4
  for (int j = 0; j < PADLEN / 64; ++j) sc[r][ix + 64 * j] *= rinv;
  __syncthreads();

  // ---- Phase 3: out[4, 128] = probs @ V. Wave w owns 16 head-dims. ----
  const float* vbase    = cache_v + ((long)(b * NKVv + g)) * SWv * HDv;
  const float* vrow_new = vnew + (long)(b * NKVv + g) * HDv;
  {
    const int n3 = wave * 16 + (lane & 15);
    const int pr = lane & 3;                 // clamped prob row
    v8f acc = {};
    int soff = k0;                           // column in sc row
    int voff = k0 * HDv + n3;                // element offset in vbase
    // Main loop: rows 0..2043, pure induction, branch-free.
#pragma unroll 4
    for (int kb = 0; kb < PADLEN - 4; kb += 4) {
      v2f a = *(const v2f*)&sc[pr][soff];
      v2f bb;
      bb[0] = vbase[voff];
      bb[1] = vbase[voff + HDv];
      soff += 4;
      voff += 4 * HDv;
      acc = WMMA_F32X4(acc, a, bb);
    }
    // Epilogue: rows 2044..2047; row 2047 comes from the fresh v.
    {
      v2f a = *(const v2f*)&sc[pr][2044 + k0];
      v2f bb;
      bb[0] = vbase[(2044 + k0) * HDv + n3];
      bb[1] = (k0 == 2) ? vrow_new[n3] : vbase[2045 * HDv + n3];
      acc = WMMA_F32X4(acc, a, bb);
    }
    if (lane < 16) {
#pragma unroll
      for (int v = 0; v < 4; ++v)
        out[((long)b * NHv + g * 4 + v) * HDv + wave * 16 + (lane & 15)] = acc[v];
    }
  }
}

// ---------------------------------------------------------------------------
// Pipeline:
//   1-3) q_raw/k_raw/v = x @ {wq,wk,wv}          (WMMA GEMM, M=8)
//   4)   q = per-batch q_raw @ rot * 1/sqrt(HD)  (WMMA GEMM, M=32, batched)
//   5)   k = per-batch k_raw @ rot               (WMMA GEMM, M=8, batched)
//   6)   fused attention with functional cache-row substitution
//   7)   dense = attn_out @ wo                   (WMMA GEMM, M=8)
// ---------------------------------------------------------------------------
extern "C" void kernel_launch(void* const* d_in, const int* in_sizes, int n_in,
                              void* d_out, int out_size, void* d_ws, size_t ws_size,
                              hipStream_t stream) {
  (void)in_sizes; (void)n_in; (void)out_size; (void)ws_size;
  const float* x    = (const float*)d_in[0];  // [8, 4096]
  const float* wq   = (const float*)d_in[1];  // [4096, 4096]
  const float* wk   = (const float*)d_in[2];  // [4096, 1024]
  const float* wv   = (const float*)d_in[3];  // [4096, 1024]
  const float* wo   = (const float*)d_in[4];  // [4096, 4096]
  const float* rot  = (const float*)d_in[5];  // [8, 128, 128]
  const float* ck   = (const float*)d_in[6];  // [8, 8, 4096, 128]
  const float* cv   = (const float*)d_in[7];  // [8, 8, 4096, 128]
  const float* mask = (const float*)d_in[8];  // [32, 8, 2048]
  float* outp = (float*)d_out;                // [8, 4096]

  float* ws     = (float*)d_ws;
  float* q_raw  = ws;             // 8*4096 = 32768
  float* k_raw  = ws + 32768;     // 8*1024 =  8192
  float* v_new  = ws + 40960;     // 8*1024 =  8192
  float* q_rot  = ws + 49152;     // 32768
  float* k_rot  = ws + 81920;     //  8192
  float* attn_o = ws + 90112;     // 32768   (total ~480KB)

  const dim3 wblk(32);
  const float qscale = 0.08838834764831845f;  // 1/sqrt(128)

  gemm16_f32_wmma<<<dim3(DIMv/16, 1, 1), wblk, 0, stream>>>(x, wq, q_raw, Bv, DIMv, DIMv, 0, 0, 0, 1.0f);
  gemm16_f32_wmma<<<dim3(NKVv*HDv/16, 1, 1), wblk, 0, stream>>>(x, wk, k_raw, Bv, NKVv*HDv, DIMv, 0, 0, 0, 1.0f);
  gemm16_f32_wmma<<<dim3(NKVv*HDv/16, 1, 1), wblk, 0, stream>>>(x, wv, v_new, Bv, NKVv*HDv, DIMv, 0, 0, 0, 1.0f);

  // RoPE: per batch b, (heads x 128) @ rot_b (128x128); score scale folded into q.
  gemm16_f32_wmma<<<dim3(HDv/16, 2, Bv), wblk, 0, stream>>>(q_raw, rot, q_rot,
      NHv, HDv, HDv, (long)NHv*HDv, (long)HDv*HDv, (long)NHv*HDv, qscale);
  gemm16_f32_wmma<<<dim3(HDv/16, 1, Bv), wblk, 0, stream>>>(k_raw, rot, k_rot,
      NKVv, HDv, HDv, (long)NKVv*HDv, (long)HDv*HDv, (long)NKVv*HDv, 1.0f);

  attn_decode<<<dim3(Bv * NKVv), dim3(256), 0, stream>>>(q_rot, k_rot, v_new, ck, cv, mask, attn_o);

  gemm16_f32_wmma<<<dim3(DIMv/16, 1, 1), wblk, 0, stream>>>(attn_o, wo, outp, Bv, DIMv, DIMv, 0, 0, 0, 1.0f);
}